// LowRankChristoffel_3822520893670
// MI455X (gfx1250) — compile-verified
//
#include <hip/hip_runtime.h>
#include <hip/hip_bf16.h>
#include <stdint.h>

typedef float v2f __attribute__((ext_vector_type(2)));
typedef float v8f __attribute__((ext_vector_type(8)));

#define D_DIM   2048
#define R_DIM   16
#define TT      16          // tokens per tile (WMMA M)
#define NW      8           // waves per block (256 threads, wave32)
#define CH      512         // d-chunk staged in LDS per iteration
#define CHP     (CH + 4)    // padded LDS row -> conflict-free b64 A-fragment reads
#define NCH     (D_DIM / CH)
#define CLAMP_V 5.0f

__global__ __launch_bounds__(256) void lowrank_christoffel_kernel(
    const float* __restrict__ v,
    const float* __restrict__ x,
    const float* __restrict__ U,
    const float* __restrict__ W,
    const float* __restrict__ Vw,
    float* __restrict__ out)
{
    __shared__ float vrow[TT][CHP];        // ~33 KB  staged v tile chunk
    __shared__ float red[NW][32][8];       //  8 KB  cross-wave C reduction
    __shared__ float sq_lds[TT * 17];      //  sq = proj^2, padded rows
    __shared__ float smod[TT];             //  sigmoid modulation per token

    const int tid   = threadIdx.x;
    const int lane  = tid & 31;
    const int wv    = tid >> 5;
    const int m     = lane & 15;            // M (A) / N (B,D) index in fragment
    const int koff  = (lane >> 4) << 1;     // 0 for lanes 0-15, 2 for lanes 16-31
    const long tokBase = (long)blockIdx.x * TT;

    // ---------- Phase 0: modulation = sigmoid(x . Vw) per token ----------
    {
        const float4* x4  = (const float4*)x;
        const float4* Vw4 = (const float4*)Vw;
        for (int tt = 0; tt < 2; ++tt) {
            const int  t   = wv * 2 + tt;
            const long row = (tokBase + t) * (D_DIM / 4);
            float acc = 0.0f;
#pragma unroll 4
            for (int i = 0; i < (D_DIM / 4) / 32; ++i) {
                const int idx = lane + i * 32;           // coalesced 512B/instr
                float4 xv = x4[row + idx];
                float4 wv4 = Vw4[idx];
                acc += xv.x * wv4.x + xv.y * wv4.y + xv.z * wv4.z + xv.w * wv4.w;
            }
            for (int off = 16; off > 0; off >>= 1)
                acc += __shfl_xor(acc, off, 32);
            if (lane == 0) smod[t] = 1.0f / (1.0f + __expf(-acc));
        }
    }

    // ---------- Phase 1: proj = v @ U  (16 tokens x 16 ranks) ----------
    v8f c;
#pragma unroll
    for (int j = 0; j < 8; ++j) c[j] = 0.0f;

    const uint64_t vbase = (uint64_t)(uintptr_t)v;

    for (int ch = 0; ch < NCH; ++ch) {
        __syncthreads();   // previous chunk's LDS reads done before overwrite
        // Stage v[tile, ch*CH : ch*CH+CH] -> LDS via async copy (ASYNCcnt path):
        // per lane: 16B from global straight into LDS, no VGPR round-trip.
        {
#pragma unroll
            for (int i = 0; i < (TT * CH / 4) / 256; ++i) {
                const int g  = tid + i * 256;
                const int t  = g >> 7;      // / (CH/4)
                const int dq = g & 127;     // float4 index within row chunk
                const uint64_t gaddr = vbase +
                    (((uint64_t)(tokBase + t) * D_DIM) + (uint64_t)(ch * CH + dq * 4)) * 4ull;
                const uint32_t laddr = (uint32_t)(uintptr_t)&vrow[t][dq * 4];
                asm volatile("global_load_async_to_lds_b128 %0, %1, off"
                             :: "v"(laddr), "v"(gaddr) : "memory");
            }
            asm volatile("s_wait_asynccnt 0x0" ::: "memory");
        }
        __syncthreads();

        // Each wave reduces its 64-d slice: 16 WMMA K-steps of 4
        const int dbase_c = wv * (CH / NW);
#pragma unroll
        for (int k = 0; k < (CH / NW) / 4; ++k) {
            const int d0c = dbase_c + k * 4;       // d within chunk
            const int d0  = ch * CH + d0c;         // global d
            v2f a, b;
            // A: 16x4 f32 fragment from LDS (conflict-free, contiguous pair)
            a.x = vrow[m][d0c + koff];
            a.y = vrow[m][d0c + koff + 1];
            // B: 4x16 fragment of U (row-major [D,R]); lanes contiguous in r
            b.x = U[(d0 + koff)     * R_DIM + m];
            b.y = U[(d0 + koff + 1) * R_DIM + m];
            c = __builtin_amdgcn_wmma_f32_16x16x4_f32(
                    false, a, false, b, (short)0, c, false, false);
        }
    }

    // Cross-wave reduction of partial proj accumulators, then square
#pragma unroll
    for (int j = 0; j < 8; ++j) red[wv][lane][j] = c[j];
    __syncthreads();
    {
        const int M    = tid >> 4;
        const int N    = tid & 15;
        const int lsrc = (M < 8) ? N : (N + 16);
        const int j    = M & 7;
        float s = 0.0f;
#pragma unroll
        for (int w = 0; w < NW; ++w) s += red[w][lsrc][j];
        sq_lds[M * 17 + N] = s * s;
    }
    __syncthreads();

    // ---------- Phase 2: out = sq @ W^T, modulate, clamp, store ----------
    const int dW  = wv * (D_DIM / NW);     // this wave's 256-wide d slice
    const int hlf = (lane >> 4) << 3;      // 0 or 8 (C/D row offset)
    for (int g = 0; g < (D_DIM / NW) / 16; ++g) {
        const int d0 = dW + g * 16;
        v8f c2;
#pragma unroll
        for (int j = 0; j < 8; ++j) c2[j] = 0.0f;
#pragma unroll
        for (int r0 = 0; r0 < R_DIM; r0 += 4) {
            v2f a, b;
            // A: sq (16 tokens x 16 ranks), K = rank
            a.x = sq_lds[m * 17 + r0 + koff];
            a.y = sq_lds[m * 17 + r0 + koff + 1];
            // B[K=r][N=dcol] = W[d0+N, r] ; contiguous 64-bit load per lane
            b.x = W[(d0 + m) * R_DIM + r0 + koff];
            b.y = W[(d0 + m) * R_DIM + r0 + koff + 1];
            c2 = __builtin_amdgcn_wmma_f32_16x16x4_f32(
                     false, a, false, b, (short)0, c2, false, false);
        }
        // Epilogue: modulation + clamp + coalesced store
#pragma unroll
        for (int j = 0; j < 8; ++j) {
            const int M = j + hlf;
            float o = c2[j] * (1.0f + smod[M]);
            o = fminf(fmaxf(o, -CLAMP_V), CLAMP_V);
            out[(tokBase + M) * D_DIM + d0 + m] = o;
        }
    }
}

extern "C" void kernel_launch(void* const* d_in, const int* in_sizes, int n_in,
                              void* d_out, int out_size, void* d_ws, size_t ws_size,
                              hipStream_t stream) {
    const float* v  = (const float*)d_in[0];
    const float* x  = (const float*)d_in[1];
    const float* U  = (const float*)d_in[2];
    const float* W  = (const float*)d_in[3];
    const float* Vw = (const float*)d_in[4];
    float* out = (float*)d_out;

    const int tokens = 4 * 4096;           // B * S
    dim3 grid(tokens / TT);                // 1024 tiles, exact
    lowrank_christoffel_kernel<<<grid, 256, 0, stream>>>(v, x, U, W, Vw, out);
    (void)in_sizes; (void)n_in; (void)out_size; (void)d_ws; (void)ws_size;
}